// LossFunction_LNCC_58445914964250
// MI455X (gfx1250) — compile-verified
//
#include <hip/hip_runtime.h>
#include <hip/hip_bf16.h>

// ---------------------------------------------------------------------------
// CDNA5 / gfx1250 implementation of the LNCC multi-scale registration loss.
//
// Box-filter sums (the core of the NCC terms) are computed as banded matrix
// products on the WMMA unit:
//   V(16x32) = Mv(16x32) * X(32x32)      (vertical k-tap sum, haloed region)
//   S(16x16) = V(16x32) * Mh(32x16)      (horizontal k-tap sum)
// decomposed into V_WMMA_F32_16X16X4_F32 K-chunks of 4. One wave32 per tile.
//
// Refinements vs round 0 (driven by the instruction histogram):
//  * band indicators precomputed once into 8 v2f regs (shared by stage-1 A
//    and stage-2 B) -> removes per-WMMA v_cmp/v_cndmask chains.
//  * stage 1 runs K-chunk outer / 5 map accumulators inner -> each LDS region
//    value is loaded once and feeds 5 WMMAs; accumulator RAW distance = 5.
//  * X regions stored column-major, V rows row-major, both with even stride
//    34 -> operand pairs are contiguous + 8B aligned -> ds_load_b64.
//
// Layout assumptions (CDNA5 ISA 7.12.2, compile-only environment):
//   A 16x4 f32 : lanes 0-15 -> {A[m][0],A[m][1]}, lanes 16-31 -> {A[m][2],A[m][3]}
//   B 4x16 f32 : lanes 0-15 -> {B[0][n],B[1][n]}, lanes 16-31 -> {B[2][n],B[3][n]}
//   C/D 16x16  : vgpr v, lane L -> D[(L<16 ? v : v+8)][L%16]
// Cross-lane data is routed through LDS so only these per-lane maps matter.
// ---------------------------------------------------------------------------

typedef __attribute__((ext_vector_type(2))) float v2f;
typedef __attribute__((ext_vector_type(8))) float v8f;

#define WAVES_PER_BLOCK 2
#define XSTRIDE   34                       // even -> 8B-aligned v2f pairs
#define XREG      (32 * XSTRIDE)           // one 32x32 region (column-major)
#define VMAP      (16 * XSTRIDE)           // one 16x32 V tile (row-major)
#define LDS_PER_WAVE (2 * XREG + 5 * VMAP) // Xi + Xj + V[5] = 4896 floats

__device__ __forceinline__ float wave_reduce_add(float v) {
    #pragma unroll
    for (int off = 16; off > 0; off >>= 1) v += __shfl_down(v, off, 32);
    return v;
}

// --------------------------- NCC tile kernel -------------------------------
// One wave computes one 16x16 output tile: 5 box-filtered maps via WMMA,
// then cc accumulation; block writes one deterministic partial sum.
template <int K>
__global__ __launch_bounds__(WAVES_PER_BLOCK * 32)
void ncc_tile_kernel(const float* __restrict__ I, const float* __restrict__ J,
                     int B, int H, int W, float* __restrict__ partial)
{
    constexpr int HALO = K / 2;
    __shared__ float lds[WAVES_PER_BLOCK * LDS_PER_WAVE];
    __shared__ float wsum[WAVES_PER_BLOCK];

    const int lane = threadIdx.x & 31;
    const int wid  = threadIdx.x >> 5;
    float* Xi = &lds[wid * LDS_PER_WAVE];   // 32 cols x 32 rows, column-major
    float* Xj = Xi + XREG;
    float* Vm = Xj + XREG;                  // 5 maps x (16 rows x 32 cols)

    const int tx   = W >> 4;
    const int tpi  = tx * (H >> 4);
    const int tile = blockIdx.x * WAVES_PER_BLOCK + wid;
    const int b    = tile / tpi;
    const int rem  = tile - b * tpi;
    const int tr   = (rem / tx) << 4;
    const int tc   = (rem % tx) << 4;
    const float* Ib = I + (size_t)b * H * W;
    const float* Jb = J + (size_t)b * H * W;

    const int  mrow = lane & 15;
    const bool hi   = lane >= 16;

    // Banded-ones operand, shared by stage-1 A and stage-2 B:
    //   band[kc] = { ind(mrow, j0), ind(mrow, j0+1) },  ind(m,j) = j in [m, m+K-1]
    v2f band[8];
    #pragma unroll
    for (int kc = 0; kc < 8; ++kc) {
        const int j0 = 4 * kc + (hi ? 2 : 0);
        band[kc].x = (j0     >= mrow && j0     < mrow + K) ? 1.0f : 0.0f;
        band[kc].y = (j0 + 1 >= mrow && j0 + 1 < mrow + K) ? 1.0f : 0.0f;
    }

    // Cooperative 32x32 haloed region load; zero outside image (conv zero-pad).
    // Column-major store: lane owns column `lane`; stride-34 banking is
    // conflict-free (17L mod 32 is a permutation).
    {
        const int gc = tc - HALO + lane;
        const bool cok = (gc >= 0) & (gc < W);
        #pragma unroll 4
        for (int r = 0; r < 32; ++r) {
            const int gr = tr - HALO + r;
            const bool ok = cok & (gr >= 0) & (gr < H);
            Xi[lane * XSTRIDE + r] = ok ? Ib[gr * W + gc] : 0.0f;
            Xj[lane * XSTRIDE + r] = ok ? Jb[gr * W + gc] : 0.0f;
        }
    }
    asm volatile("s_wait_dscnt 0" ::: "memory");

    // ---- stage 1: V_m = Mv * X_m for the 5 maps, two 16-column tiles ----
    const int col = mrow;  // within-tile output column this lane supplies
    #pragma unroll
    for (int ct = 0; ct < 2; ++ct) {
        v8f acc[5];
        #pragma unroll
        for (int m = 0; m < 5; ++m) acc[m] = (v8f){};
        const int c = ct * 16 + col;
        const float* xip = &Xi[c * XSTRIDE];
        const float* xjp = &Xj[c * XSTRIDE];
        #pragma unroll
        for (int kc = 0; kc < 8; ++kc) {
            const int j0 = 4 * kc + (hi ? 2 : 0);
            const v2f xi = *(const v2f*)&xip[j0];   // ds_load_b64
            const v2f xj = *(const v2f*)&xjp[j0];
            acc[0] = __builtin_amdgcn_wmma_f32_16x16x4_f32(
                         false, band[kc], false, xi,      (short)0, acc[0], false, false);
            acc[1] = __builtin_amdgcn_wmma_f32_16x16x4_f32(
                         false, band[kc], false, xj,      (short)0, acc[1], false, false);
            acc[2] = __builtin_amdgcn_wmma_f32_16x16x4_f32(
                         false, band[kc], false, xi * xi, (short)0, acc[2], false, false);
            acc[3] = __builtin_amdgcn_wmma_f32_16x16x4_f32(
                         false, band[kc], false, xj * xj, (short)0, acc[3], false, false);
            acc[4] = __builtin_amdgcn_wmma_f32_16x16x4_f32(
                         false, band[kc], false, xi * xj, (short)0, acc[4], false, false);
        }
        // D layout -> LDS V buffers (row-major, stride 34)
        #pragma unroll
        for (int m = 0; m < 5; ++m)
            #pragma unroll
            for (int v = 0; v < 8; ++v)
                Vm[m * VMAP + (hi ? v + 8 : v) * XSTRIDE + c] = acc[m][v];
    }
    asm volatile("s_wait_dscnt 0" ::: "memory");

    // ---- stage 2: S_m = V_m * Mh ----
    v8f S[5];
    #pragma unroll
    for (int m = 0; m < 5; ++m) {
        v8f c = {};
        const float* vp = &Vm[m * VMAP + mrow * XSTRIDE];
        #pragma unroll
        for (int kc = 0; kc < 8; ++kc) {
            const int j0 = 4 * kc + (hi ? 2 : 0);
            const v2f a = *(const v2f*)&vp[j0];     // ds_load_b64
            c = __builtin_amdgcn_wmma_f32_16x16x4_f32(
                    false, a, false, band[kc], (short)0, c, false, false);
        }
        S[m] = c;
    }

    // cc per pixel, sum over the 8 pixels this lane holds.
    const float win = (float)(K * K);
    float acc = 0.0f;
    #pragma unroll
    for (int v = 0; v < 8; ++v) {
        const float Is = S[0][v], Js = S[1][v];
        const float I2 = S[2][v], J2 = S[3][v], IJ = S[4][v];
        const float cross = IJ - Is * Js / win;
        const float Iv = I2 - Is * Is / win;
        const float Jv = J2 - Js * Js / win;
        acc += cross * cross / (Iv * Jv + 1e-5f);
    }
    acc = wave_reduce_add(acc);
    if (lane == 0) wsum[wid] = acc;
    __syncthreads();
    if (threadIdx.x == 0) {
        float s = 0.0f;
        for (int i = 0; i < WAVES_PER_BLOCK; ++i) s += wsum[i];
        partial[blockIdx.x] = s;   // deterministic per-block partial
    }
}

// --------------------------- resize (align-corners bilinear * factor) ------
__global__ __launch_bounds__(256)
void resize_kernel(const float* __restrict__ in, float* __restrict__ out,
                   int B, int H, int W, int oh, int ow, float factor)
{
    const int idx = blockIdx.x * blockDim.x + threadIdx.x;
    const int total = B * oh * ow;
    if (idx >= total) return;
    const int x = idx % ow;
    const int t = idx / ow;
    const int y = t % oh;
    const int b = t / oh;
    const float py = (float)(y * (H - 1)) / (float)(oh - 1);
    const float px = (float)(x * (W - 1)) / (float)(ow - 1);
    const int y0 = (int)floorf(py), x0 = (int)floorf(px);
    const int y1 = min(y0 + 1, H - 1), x1 = min(x0 + 1, W - 1);
    const float wy = py - (float)y0, wx = px - (float)x0;
    const float* p = in + (size_t)b * H * W;
    const float v00 = p[y0 * W + x0], v01 = p[y0 * W + x1];
    const float v10 = p[y1 * W + x0], v11 = p[y1 * W + x1];
    const float vy0 = v00 * (1.0f - wy) + v10 * wy;   // y first (matches ref)
    const float vy1 = v01 * (1.0f - wy) + v11 * wy;
    out[idx] = factor * (vy0 * (1.0f - wx) + vy1 * wx);
}

// --------------------------- warp (bilinear, zero outside) -----------------
__global__ __launch_bounds__(256)
void warp_kernel(const float* __restrict__ src, const float* __restrict__ flow,
                 float* __restrict__ out, int B, int H, int W)
{
    const int idx = blockIdx.x * blockDim.x + threadIdx.x;
    const int total = B * H * W;
    if (idx >= total) return;
    const int x = idx % W;
    const int t = idx / W;
    const int y = t % H;
    const int b = t / H;
    const float* f = flow + (size_t)b * 2 * H * W;
    const float yy = (float)y + f[y * W + x];
    const float xx = (float)x + f[H * W + y * W + x];
    const float y0f = floorf(yy), x0f = floorf(xx);
    const float wy = yy - y0f, wx = xx - x0f;
    const int y0 = (int)y0f, x0 = (int)x0f;
    const float* p = src + (size_t)b * H * W;
    auto tap = [&](int yi, int xi) -> float {
        const int yc = min(max(yi, 0), H - 1);
        const int xc = min(max(xi, 0), W - 1);
        const float v = p[yc * W + xc];
        const bool valid = (yi >= 0) & (yi < H) & (xi >= 0) & (xi < W);
        return valid ? v : 0.0f;
    };
    out[idx] = tap(y0, x0) * (1.0f - wy) * (1.0f - wx)
             + tap(y0, x0 + 1) * (1.0f - wy) * wx
             + tap(y0 + 1, x0) * wy * (1.0f - wx)
             + tap(y0 + 1, x0 + 1) * wy * wx;
}

// --------------------------- gradient loss partials ------------------------
__global__ __launch_bounds__(256)
void grad_kernel(const float* __restrict__ flow, int B2, int H, int W,
                 float* __restrict__ pdx, float* __restrict__ pdy)
{
    __shared__ float sdx[8], sdy[8];
    float adx = 0.0f, ady = 0.0f;
    const int stride = gridDim.x * blockDim.x;
    const int ndx = B2 * H * (W - 1);
    for (int i = blockIdx.x * blockDim.x + threadIdx.x; i < ndx; i += stride) {
        const int w = i % (W - 1);
        const int t = i / (W - 1);
        const int h = t % H;
        const int c = t / H;
        const size_t base = (size_t)c * H * W + (size_t)h * W + w;
        const float d = flow[base + 1] - flow[base];
        adx += d * d;
    }
    const int ndy = B2 * (H - 1) * W;
    for (int i = blockIdx.x * blockDim.x + threadIdx.x; i < ndy; i += stride) {
        const int w = i % W;
        const int t = i / W;
        const int h = t % (H - 1);
        const int c = t / (H - 1);
        const size_t base = (size_t)c * H * W + (size_t)h * W + w;
        const float d = flow[base + W] - flow[base];
        ady += d * d;
    }
    adx = wave_reduce_add(adx);
    ady = wave_reduce_add(ady);
    const int lane = threadIdx.x & 31, wid = threadIdx.x >> 5;
    if (lane == 0) { sdx[wid] = adx; sdy[wid] = ady; }
    __syncthreads();
    if (threadIdx.x == 0) {
        float a = 0.0f, c = 0.0f;
        for (int i = 0; i < (int)(blockDim.x >> 5); ++i) { a += sdx[i]; c += sdy[i]; }
        pdx[blockIdx.x] = a;
        pdy[blockIdx.x] = c;
    }
}

// --------------------------- finalize --------------------------------------
__device__ float block_sum_array(const float* __restrict__ p, int n, float* sh)
{
    float a = 0.0f;
    for (int i = threadIdx.x; i < n; i += blockDim.x) a += p[i];
    a = wave_reduce_add(a);
    const int lane = threadIdx.x & 31, wid = threadIdx.x >> 5;
    __syncthreads();
    if (lane == 0) sh[wid] = a;
    __syncthreads();
    float r = 0.0f;
    for (int i = 0; i < (int)(blockDim.x >> 5); ++i) r += sh[i];
    return r;
}

__global__ __launch_bounds__(256)
void finalize_kernel(const float* p0, int n0, const float* p1, int n1,
                     const float* p2, int n2, const float* pdx,
                     const float* pdy, int ng, float* __restrict__ out)
{
    __shared__ float sh[8];
    const float s0  = block_sum_array(p0, n0, sh);
    const float s1  = block_sum_array(p1, n1, sh);
    const float s2  = block_sum_array(p2, n2, sh);
    const float sgx = block_sum_array(pdx, ng, sh);
    const float sgy = block_sum_array(pdy, ng, sh);
    if (threadIdx.x == 0) {
        const float ncc  = -(s0 / (64.0f * 256.0f * 256.0f));
        const float m1   = -(s1 / (64.0f * 128.0f * 128.0f));
        const float m2   = -(s2 / (64.0f * 64.0f * 64.0f));
        const float grad = 0.5f * (sgx / 8355840.0f + sgy / 8355840.0f);
        const float loss = 3.2f * m1 + 0.8f * m2 + 10.0f * ncc + 15.0f * grad;
        out[0] = loss;
        out[1] = ncc;
        out[2] = grad;
    }
}

// --------------------------- host entry ------------------------------------
extern "C" void kernel_launch(void* const* d_in, const int* in_sizes, int n_in,
                              void* d_out, int out_size, void* d_ws, size_t ws_size,
                              hipStream_t stream)
{
    const float* y     = (const float*)d_in[0];   // (64,1,256,256)
    const float* tgt   = (const float*)d_in[1];   // (64,1,256,256)
    const float* src   = (const float*)d_in[2];   // (64,1,256,256)
    const float* flow  = (const float*)d_in[3];   // (64,2,256,256)
    const float* flow1 = (const float*)d_in[4];   // (64,2,128,128)
    const float* flow2 = (const float*)d_in[5];   // (64,2,64,64)
    float* out = (float*)d_out;                   // (loss, ncc, grad)
    float* ws  = (float*)d_ws;

    const int B = 64, H = 256, W = 256;
    const int N1 = 64 * 128 * 128;   // half-res plane count
    const int N2 = 64 * 64 * 64;     // quarter-res plane count
    const int NB0 = 16384 / WAVES_PER_BLOCK;  // NCC blocks, scale 0
    const int NB1 = 4096 / WAVES_PER_BLOCK;   // NCC blocks, scale 1
    const int NB2 = 1024 / WAVES_PER_BLOCK;   // NCC blocks, scale 2

    float* z1t = ws;            // half-res tgt * 0.5
    float* z1s = z1t + N1;      // half-res src * 0.5
    float* w1  = z1s + N1;      // warp(z1s, flow1)
    float* z2t = w1 + N1;       // quarter-res tgt * 0.25
    float* z2s = z2t + N2;      // quarter-res src * 0.25
    float* w2  = z2s + N2;      // warp(z2s, flow2)
    float* p0  = w2 + N2;       // NCC partials (scale 0, k=15)
    float* p1  = p0 + NB0;      // NCC partials (scale 1, k=7)
    float* p2  = p1 + NB1;      // NCC partials (scale 2, k=5)
    float* pdx = p2 + NB2;      // 512 grad-x partials
    float* pdy = pdx + 512;     // 512 grad-y partials

    // Multi-scale pyramid (factor-scaled align-corners bilinear).
    resize_kernel<<<(N1 + 255) / 256, 256, 0, stream>>>(tgt, z1t, B, H, W, 128, 128, 0.5f);
    resize_kernel<<<(N1 + 255) / 256, 256, 0, stream>>>(src, z1s, B, H, W, 128, 128, 0.5f);
    resize_kernel<<<(N2 + 255) / 256, 256, 0, stream>>>(tgt, z2t, B, H, W, 64, 64, 0.25f);
    resize_kernel<<<(N2 + 255) / 256, 256, 0, stream>>>(src, z2s, B, H, W, 64, 64, 0.25f);

    // Warps.
    warp_kernel<<<(N1 + 255) / 256, 256, 0, stream>>>(z1s, flow1, w1, B, 128, 128);
    warp_kernel<<<(N2 + 255) / 256, 256, 0, stream>>>(z2s, flow2, w2, B, 64, 64);

    // WMMA box-filter NCC at each scale. One wave32 per 16x16 tile.
    ncc_tile_kernel<15><<<NB0, WAVES_PER_BLOCK * 32, 0, stream>>>(tgt, y, B, 256, 256, p0);
    ncc_tile_kernel< 7><<<NB1, WAVES_PER_BLOCK * 32, 0, stream>>>(w1, z1t, B, 128, 128, p1);
    ncc_tile_kernel< 5><<<NB2, WAVES_PER_BLOCK * 32, 0, stream>>>(w2, z2t, B, 64, 64, p2);

    // Gradient loss partials over flow (treated as 128 planes of 256x256).
    grad_kernel<<<512, 256, 0, stream>>>(flow, 128, 256, 256, pdx, pdy);

    // Final deterministic combine.
    finalize_kernel<<<1, 256, 0, stream>>>(p0, NB0, p1, NB1, p2, NB2, pdx, pdy, 512, out);
}